// StandNet_40415642255439
// MI455X (gfx1250) — compile-verified
//
#include <hip/hip_runtime.h>
#include <math.h>

typedef __attribute__((ext_vector_type(2))) float v2f;
typedef __attribute__((ext_vector_type(8))) float v8f;

#define KCOLS 1024
#define TILE_ROWS 16

// One wave (32 lanes) per 16-row tile. Pure streaming + WMMA cross-lane reduce.
__global__ __launch_bounds__(32) void rownorm_l2_kernel(const float* __restrict__ X,
                                                        float* __restrict__ Y,
                                                        int rows) {
    const int lane = threadIdx.x;            // 0..31 (wave32)
    const long r0  = (long)blockIdx.x * TILE_ROWS;

    __shared__ float Pl[TILE_ROWS * 32];     // partial sums: [m*32 + lane]
    __shared__ float Sl[TILE_ROWS];          // per-row sum of squares

    const float* __restrict__ xb = X + r0 * KCOLS;
    float*       __restrict__ yb = Y + r0 * KCOLS;

    // ---- Pass 1: coalesced accumulate of squares (float4 = global_load_b128) ----
    float acc[TILE_ROWS];
#pragma unroll
    for (int m = 0; m < TILE_ROWS; ++m) acc[m] = 0.0f;

#pragma unroll
    for (int m = 0; m < TILE_ROWS; ++m) {
        if (r0 + m < rows) {                                   // uniform per wave
            const float4* rp = (const float4*)(xb + (long)m * KCOLS);
            float a = 0.0f;
#pragma unroll
            for (int j = 0; j < KCOLS / 128; ++j) {
                float4 v = rp[lane + 32 * j];                  // col = 4*lane + 128*j
                a = fmaf(v.x, v.x, a);
                a = fmaf(v.y, v.y, a);
                a = fmaf(v.z, v.z, a);
                a = fmaf(v.w, v.w, a);
            }
            acc[m] = a;
        }
    }

    // Stage the 16x32 partial matrix P in LDS.
#pragma unroll
    for (int m = 0; m < TILE_ROWS; ++m) Pl[m * 32 + lane] = acc[m];
    __syncthreads();

    // ---- Cross-lane reduce with V_WMMA_F32_16X16X4_F32: D = P_chunk * ones + C ----
    // A layout (ISA 16x4 f32): lanes 0-15 hold M=lane, VGPR0/1 = K0/K1;
    // lanes 16-31 hold M=lane-16, K2/K3. With B == all-ones, D[m][*] = sum_k P[m][k].
    v8f c = {};
    const v2f bones = {1.0f, 1.0f};
    const int mrow  = lane & 15;
    const int khalf = (lane >> 4) << 1;                        // 0 or 2
#pragma unroll
    for (int k0 = 0; k0 < 32; k0 += 4) {
        v2f a;
        a.x = Pl[mrow * 32 + k0 + khalf + 0];
        a.y = Pl[mrow * 32 + k0 + khalf + 1];
        c = __builtin_amdgcn_wmma_f32_16x16x4_f32(
                /*neg_a=*/false, a, /*neg_b=*/false, bones,
                /*c_mod=*/(short)0, c, /*reuse_a=*/false, /*reuse_b=*/false);
    }

    // D layout (16x16 f32 C/D): lanes 0-15: N=lane, VGPR v => M=v;
    // lanes 16-31: N=lane-16, VGPR v => M=v+8. All columns identical here.
    if (lane == 0) {
#pragma unroll
        for (int v = 0; v < 8; ++v) Sl[v] = c[v];
    }
    if (lane == 16) {
#pragma unroll
        for (int v = 0; v < 8; ++v) Sl[8 + v] = c[v];
    }
    __syncthreads();

    // ---- Pass 2: scale rows; re-read is an L2 hit (192 MB L2, tight reuse) ----
#pragma unroll
    for (int m = 0; m < TILE_ROWS; ++m) {
        if (r0 + m < rows) {                                   // uniform per wave
            const float s = Sl[m];
            const float d = (s > 0.0f) ? (1.0f / sqrtf(s)) : 0.0f;  // safe reciprocal
            const float4* rp = (const float4*)(xb + (long)m * KCOLS);
            float4*       wp = (float4*)(yb + (long)m * KCOLS);
#pragma unroll
            for (int j = 0; j < KCOLS / 128; ++j) {
                float4 v = rp[lane + 32 * j];
                v.x *= d; v.y *= d; v.z *= d; v.w *= d;
                wp[lane + 32 * j] = v;
            }
        }
    }
}

extern "C" void kernel_launch(void* const* d_in, const int* in_sizes, int n_in,
                              void* d_out, int out_size, void* d_ws, size_t ws_size,
                              hipStream_t stream) {
    (void)n_in; (void)out_size; (void)d_ws; (void)ws_size;
    const float* X = (const float*)d_in[0];
    float*       Y = (float*)d_out;

    const int rows  = in_sizes[0] / KCOLS;          // 32768 for the reference
    const int tiles = (rows + TILE_ROWS - 1) / TILE_ROWS;

    rownorm_l2_kernel<<<tiles, 32, 0, stream>>>(X, Y, rows);
}